// ReviewAggregatorConv_11433202942499
// MI455X (gfx1250) — compile-verified
//
#include <hip/hip_runtime.h>
#include <hip/hip_bf16.h>

#define IN_DIM 64
#define ATT_DIM 32

typedef __attribute__((ext_vector_type(2))) float v2f;
typedef __attribute__((ext_vector_type(8))) float v8f;

// ---------------------------------------------------------------------------
// K1: fold the two linear layers: w_eff = W_o @ W_att  (64), c = b_o@W_att + b_att
// ---------------------------------------------------------------------------
__global__ void k_weff(const float* __restrict__ W_o, const float* __restrict__ b_o,
                       const float* __restrict__ W_att, const float* __restrict__ b_att,
                       float* __restrict__ weff /* 65 floats: 64 weights + bias */) {
    int i = threadIdx.x;
    if (i < IN_DIM) {
        float acc = 0.f;
        #pragma unroll
        for (int k = 0; k < ATT_DIM; ++k) acc += W_o[i * ATT_DIM + k] * W_att[k];
        weff[i] = acc;
    }
    if (i == 0) {
        float c = b_att[0];
        #pragma unroll
        for (int k = 0; k < ATT_DIM; ++k) c += b_o[k] * W_att[k];
        weff[IN_DIM] = c;
    }
}

// ---------------------------------------------------------------------------
// K2: zero out, s1/s2; m1/m2 = -inf
// ---------------------------------------------------------------------------
__global__ void k_init(float* __restrict__ out, float* __restrict__ m1, float* __restrict__ s1,
                       float* __restrict__ m2, float* __restrict__ s2, int N) {
    int idx = blockIdx.x * blockDim.x + threadIdx.x;
    if (idx < N * IN_DIM) out[idx] = 0.f;
    if (idx < N) {
        float ninf = -__builtin_inff();
        m1[idx] = ninf; m2[idx] = ninf;
        s1[idx] = 0.f;  s2[idx] = 0.f;
    }
}

// ---------------------------------------------------------------------------
// K3: a[n] = dot(x[n], w_eff) + c using V_WMMA_F32_16X16X4_F32.
// One wave32 computes logits for 16 nodes.
//   A (16x4 f32): lanes 0-15 = rows M; VGPR r = K=r (lanes 0-15) / K=r+2 (lanes 16-31)
//   B (4x16 f32): lane%16 = col N; VGPR r = K=r (lanes 0-15) / K=r+2 (lanes 16-31)
//       -> w_eff lives only in column N=0 (lane 0: K=k0,k0+1 ; lane 16: K=k0+2,k0+3)
//   C/D (16x16 f32): VGPR r: lanes 0-15 -> M=r, N=lane ; lanes 16-31 -> M=r+8.
//       -> column N=0 = logits: lane 0 holds M=0..7, lane 16 holds M=8..15.
// Branchless B construction: unconditional contiguous weff loads (hoistable by
// the compiler, no EXEC-mask save/restore) masked by a 0/1 multiply.
// ---------------------------------------------------------------------------
__global__ void k_logits_wmma(const float* __restrict__ x, const float* __restrict__ weff,
                              float* __restrict__ a, int N) {
    int lane = threadIdx.x & 31;
    int wave = (blockIdx.x * blockDim.x + threadIdx.x) >> 5;
    int base = wave * 16;
    if (base >= N) return;                 // wave-uniform: EXEC stays all-ones for WMMA

    int mloc  = lane & 15;                 // A: row / B: column within half-wave
    int khalf = (lane >> 4) * 2;           // lanes 16-31 own the upper K pair
    float colmask = (mloc == 0) ? 1.0f : 0.0f;   // only column N=0 carries w_eff

    int node = base + mloc;
    node = node < N ? node : N - 1;        // address clamp only; no EXEC divergence
    const float* xrow = x + (size_t)node * IN_DIM + khalf;
    const float* wp   = weff + khalf;

    v8f acc = {};
    #pragma unroll
    for (int k0 = 0; k0 < IN_DIM; k0 += 4) {
        v2f A; A.x = xrow[k0]; A.y = xrow[k0 + 1];
        v2f B; B.x = wp[k0] * colmask; B.y = wp[k0 + 1] * colmask;
        acc = __builtin_amdgcn_wmma_f32_16x16x4_f32(
            /*neg_a=*/false, A, /*neg_b=*/false, B,
            /*c_mod=*/(short)0, acc, /*reuse_a=*/false, /*reuse_b=*/false);
    }

    float cb = weff[IN_DIM];
    if (mloc == 0) {                       // lanes 0 and 16 hold column N=0
        int moff = (lane >> 4) * 8;        // lane 0 -> M=0..7, lane 16 -> M=8..15
        #pragma unroll
        for (int r = 0; r < 8; ++r) {
            int n2 = base + moff + r;
            if (n2 < N) a[n2] = acc[r] + cb;
        }
    }
}

// ---------------------------------------------------------------------------
// float atomic max via int punning (valid with -inf init, IEEE ordering)
// ---------------------------------------------------------------------------
__device__ __forceinline__ void atomicMaxF(float* addr, float val) {
    if (val >= 0.f) atomicMax((int*)addr, __float_as_int(val));
    else            atomicMin((unsigned int*)addr, __float_as_uint(val));
}

// ---------------------------------------------------------------------------
// K4: per-dst segment max of a[src], both graphs in one launch
// ---------------------------------------------------------------------------
__global__ void k_edge_max(const float* __restrict__ a,
                           const int* __restrict__ src1, const int* __restrict__ dst1,
                           const int* __restrict__ src2, const int* __restrict__ dst2,
                           float* __restrict__ m1, float* __restrict__ m2, int E) {
    int i = blockIdx.x * blockDim.x + threadIdx.x;
    if (i >= 2 * E) return;
    const int* src; const int* dst; float* m; int e;
    if (i < E) { e = i;     src = src1; dst = dst1; m = m1; }
    else       { e = i - E; src = src2; dst = dst2; m = m2; }
    atomicMaxF(&m[dst[e]], a[src[e]]);
}

// ---------------------------------------------------------------------------
// K5: s[dst] += exp(a[src] - m[dst])
// ---------------------------------------------------------------------------
__global__ void k_edge_sum(const float* __restrict__ a,
                           const int* __restrict__ src1, const int* __restrict__ dst1,
                           const int* __restrict__ src2, const int* __restrict__ dst2,
                           const float* __restrict__ m1, float* __restrict__ s1,
                           const float* __restrict__ m2, float* __restrict__ s2, int E) {
    int i = blockIdx.x * blockDim.x + threadIdx.x;
    if (i >= 2 * E) return;
    const int* src; const int* dst; const float* m; float* s; int e;
    if (i < E) { e = i;     src = src1; dst = dst1; m = m1; s = s1; }
    else       { e = i - E; src = src2; dst = dst2; m = m2; s = s2; }
    int sn = src[e], dn = dst[e];
    atomicAdd(&s[dn], expf(a[sn] - m[dn]));
}

// ---------------------------------------------------------------------------
// K6: out[dst] += x[src] * (exp(a[src]-m[dst]) / s[dst]) * 0.5
// 4 threads per edge, 16 features each (float4 loads + scalar f32 atomics)
// ---------------------------------------------------------------------------
__global__ void k_edge_agg(const float* __restrict__ x, const float* __restrict__ a,
                           const int* __restrict__ src1, const int* __restrict__ dst1,
                           const int* __restrict__ src2, const int* __restrict__ dst2,
                           const float* __restrict__ m1, const float* __restrict__ s1,
                           const float* __restrict__ m2, const float* __restrict__ s2,
                           float* __restrict__ out, int E) {
    int q = blockIdx.x * blockDim.x + threadIdx.x;
    if (q >= 8 * E) return;                // 2 graphs * E edges * 4 quads
    int part = q & 3;
    int i = q >> 2;
    const int* src; const int* dst; const float* m; const float* s; int e;
    if (i < E) { e = i;     src = src1; dst = dst1; m = m1; s = s1; }
    else       { e = i - E; src = src2; dst = dst2; m = m2; s = s2; }
    int sn = src[e], dn = dst[e];
    float w = expf(a[sn] - m[dn]) / s[dn] * 0.5f;

    const float4* xin = (const float4*)(x + (size_t)sn * IN_DIM + part * 16);
    float* op = out + (size_t)dn * IN_DIM + part * 16;
    #pragma unroll
    for (int j = 0; j < 4; ++j) {
        float4 v = xin[j];
        atomicAdd(op + j * 4 + 0, v.x * w);
        atomicAdd(op + j * 4 + 1, v.y * w);
        atomicAdd(op + j * 4 + 2, v.z * w);
        atomicAdd(op + j * 4 + 3, v.w * w);
    }
}

// ---------------------------------------------------------------------------
extern "C" void kernel_launch(void* const* d_in, const int* in_sizes, int n_in,
                              void* d_out, int out_size, void* d_ws, size_t ws_size,
                              hipStream_t stream) {
    const float* x     = (const float*)d_in[0];
    const float* W_o   = (const float*)d_in[1];
    const float* b_o   = (const float*)d_in[2];
    const float* W_att = (const float*)d_in[3];
    const float* b_att = (const float*)d_in[4];
    const int*   src1  = (const int*)d_in[5];
    const int*   dst1  = (const int*)d_in[6];
    const int*   src2  = (const int*)d_in[7];
    const int*   dst2  = (const int*)d_in[8];
    float* out = (float*)d_out;

    const int N = in_sizes[0] / IN_DIM;
    const int E = in_sizes[5];

    // workspace layout: a[N] | m1[N] | s1[N] | m2[N] | s2[N] | weff[65]
    float* ws   = (float*)d_ws;
    float* a    = ws;
    float* m1   = ws + (size_t)N;
    float* s1   = ws + (size_t)2 * N;
    float* m2   = ws + (size_t)3 * N;
    float* s2   = ws + (size_t)4 * N;
    float* weff = ws + (size_t)5 * N;

    k_weff<<<1, 64, 0, stream>>>(W_o, b_o, W_att, b_att, weff);

    int initN = N * IN_DIM;
    k_init<<<(initN + 255) / 256, 256, 0, stream>>>(out, m1, s1, m2, s2, N);

    int waves = (N + 15) / 16;
    k_logits_wmma<<<(waves * 32 + 255) / 256, 256, 0, stream>>>(x, weff, a, N);

    int twoE = 2 * E;
    k_edge_max<<<(twoE + 255) / 256, 256, 0, stream>>>(a, src1, dst1, src2, dst2, m1, m2, E);
    k_edge_sum<<<(twoE + 255) / 256, 256, 0, stream>>>(a, src1, dst1, src2, dst2, m1, s1, m2, s2, E);

    int quads = 8 * E;
    k_edge_agg<<<(quads + 255) / 256, 256, 0, stream>>>(x, a, src1, dst1, src2, dst2,
                                                        m1, s1, m2, s2, out, E);
}